// cham_dist_85907935854709
// MI455X (gfx1250) — compile-verified
//
#include <hip/hip_runtime.h>
#include <hip/hip_bf16.h>

typedef __attribute__((ext_vector_type(2))) float v2f;
typedef __attribute__((ext_vector_type(8))) float v8f;

#define HW    16384
#define NV    16385   // valid points incl. single sentinel pad point
#define NP    16400   // padded to multiple of 16
#define NT    (NP/16) // 1025 16-wide tiles
#define BTOT  8       // B*T
#define BB    2
#define TT    4
#define SENTV 1000.0f
#define BIGC  1.0e9f

// ---------------------------------------------------------------------------
// Prep: build masked, padded point arrays as (x, y, z, ||p||^2) float4s,
// plus separate row-norm arrays for the final add.
// op:  output_points [BT][HW][3], rng: output_range [BT][HW]
// tgt: target [BT][4][HW]  (ch0 = range mask, ch1..3 = xyz)
// ---------------------------------------------------------------------------
__global__ void prep_kernel(const float* __restrict__ op,
                            const float* __restrict__ rng,
                            const float* __restrict__ tgt,
                            float4* __restrict__ opPts, float* __restrict__ opN,
                            float4* __restrict__ tpPts, float* __restrict__ tpN) {
    int gid = blockIdx.x * blockDim.x + threadIdx.x;
    if (gid >= BTOT * NP) return;
    int bt = gid / NP;
    int i  = gid % NP;
    float ox, oy, oz, tx, ty, tz;
    if (i < HW) {
        float r = rng[(size_t)bt * HW + i];
        const float* p = op + ((size_t)bt * HW + i) * 3;
        if (r < 0.0f) { ox = oy = oz = SENTV; }
        else          { ox = p[0]; oy = p[1]; oz = p[2]; }
        const float* tb = tgt + (size_t)bt * 4 * HW;
        float tr = tb[i];
        if (tr < 0.0f) { tx = ty = tz = SENTV; }
        else { tx = tb[HW + i]; ty = tb[2 * HW + i]; tz = tb[3 * HW + i]; }
    } else if (i == HW) {
        ox = oy = oz = tx = ty = tz = SENTV;   // appended sentinel pad point
    } else {
        ox = oy = oz = tx = ty = tz = BIGC;    // alignment padding, never wins min
    }
    float on = ox * ox + oy * oy + oz * oz;
    float tn = tx * tx + ty * ty + tz * tz;
    opPts[gid] = make_float4(ox, oy, oz, on);  // norm packed in .w == K=3 row of B
    opN[gid]   = on;
    tpPts[gid] = make_float4(tx, ty, tz, tn);
    tpN[gid]   = tn;
}

// ---------------------------------------------------------------------------
// NN kernel: one wave = one 16-row tile of A; loops over all column tiles of B
// using V_WMMA_F32_16X16X4_F32 with C = inline 0.
// A fragment carries (-2x,-2y | -2z, 1.0); B fragment carries (x,y | z,||b||^2)
// so D_ij = ||b_j||^2 - 2 a_i.b_j in a single WMMA. Running elementwise min
// over column tiles, shfl-reduce within 16-lane halves, add ||a_i||^2 at write.
// ---------------------------------------------------------------------------
__global__ void __launch_bounds__(256)
nn_kernel(const float4* __restrict__ opPts, const float* __restrict__ opN,
          const float4* __restrict__ tpPts, const float* __restrict__ tpN,
          float* __restrict__ min1, float* __restrict__ min2) {
    const int bt   = blockIdx.z;
    const int dir  = blockIdx.y;
    const int wave = threadIdx.x >> 5;
    const int lane = threadIdx.x & 31;
    const int tile = blockIdx.x * 8 + wave;
    if (tile >= NT) return;                       // wave-uniform, EXEC stays all-1

    const float4* A  = (dir == 0 ? opPts : tpPts) + (size_t)bt * NP;
    const float*  An = (dir == 0 ? opN   : tpN)   + (size_t)bt * NP;
    const float4* Bp = (dir == 0 ? tpPts : opPts) + (size_t)bt * NP;
    float* outMin    = (dir == 0 ? min1  : min2)  + (size_t)bt * NP;

    const int half = lane >> 4;   // 0: K0/K1   1: K2/K3
    const int l16  = lane & 15;

    // A fragment (16x4 f32): lanes 0-15 -> (-2x,-2y); lanes 16-31 -> (-2z, 1.0)
    const float* ap = (const float*)(A + tile * 16 + l16) + half * 2;
    v2f a;
    a.x = -2.0f * ap[0];
    a.y = half ? 1.0f : -2.0f * ap[1];

    v8f runmin;
#pragma unroll
    for (int k = 0; k < 8; ++k) runmin[k] = 3.4e38f;

    const v8f czero = {};   // SRC2 = inline constant 0

    for (int ct = 0; ct < NT; ++ct) {
        // B fragment (4x16 f32): lanes 0-15 -> (x,y); lanes 16-31 -> (z, ||b||^2)
        const float* bp = (const float*)(Bp + ct * 16 + l16) + half * 2;
        v2f b;
        b.x = bp[0];
        b.y = bp[1];
        v8f d = __builtin_amdgcn_wmma_f32_16x16x4_f32(
            /*neg_a=*/false, a, /*neg_b=*/false, b,
            /*c_mod=*/(short)0, czero, /*reuse_a=*/false, /*reuse_b=*/false);
#pragma unroll
        for (int k = 0; k < 8; ++k) runmin[k] = fminf(runmin[k], d[k]);
    }

    // min across the 16 lanes of each half-wave (xor masks 1,2,4,8 stay in-half)
#pragma unroll
    for (int k = 0; k < 8; ++k) {
        float v = runmin[k];
        v = fminf(v, __shfl_xor(v, 1, 32));
        v = fminf(v, __shfl_xor(v, 2, 32));
        v = fminf(v, __shfl_xor(v, 4, 32));
        v = fminf(v, __shfl_xor(v, 8, 32));
        runmin[k] = v;
    }

    // C/D layout: VGPR k, lanes 0-15 -> row k ; lanes 16-31 -> row k+8
    if (l16 == 0) {                               // lanes 0 and 16 write
        int rowBase = tile * 16 + half * 8;
#pragma unroll
        for (int k = 0; k < 8; ++k) {
            int row = rowBase + k;
            if (row < NV)
                outMin[row] = fmaxf(runmin[k] + An[row], 0.0f);
        }
    }
}

// ---------------------------------------------------------------------------
// Per-pair reduction: sum of min distances + count(>0), both directions.
// ---------------------------------------------------------------------------
__global__ void reduce_kernel(const float* __restrict__ min1,
                              const float* __restrict__ min2,
                              float* __restrict__ dc) {
    const int bt = blockIdx.x;
    __shared__ float s1[256], n1s[256], s2[256], n2s[256];
    float a1 = 0.f, n1 = 0.f, a2 = 0.f, n2 = 0.f;
    for (int i = threadIdx.x; i < NV; i += 256) {
        float d1 = min1[(size_t)bt * NP + i];
        a1 += d1; n1 += (d1 > 0.0f) ? 1.0f : 0.0f;
        float d2 = min2[(size_t)bt * NP + i];
        a2 += d2; n2 += (d2 > 0.0f) ? 1.0f : 0.0f;
    }
    int t = threadIdx.x;
    s1[t] = a1; n1s[t] = n1; s2[t] = a2; n2s[t] = n2;
    __syncthreads();
    for (int s = 128; s > 0; s >>= 1) {
        if (t < s) {
            s1[t] += s1[t + s]; n1s[t] += n1s[t + s];
            s2[t] += s2[t + s]; n2s[t] += n2s[t + s];
        }
        __syncthreads();
    }
    if (t == 0)
        dc[bt] = s1[0] / n1s[0] + s2[0] / n2s[0];
}

// ---------------------------------------------------------------------------
// Finalize: out[0..3] = chamf_dist_t (mean over B of reshape(T,B) rows),
//           out[4..11] = chamfer_matrix flat (== dist_combined flat).
// ---------------------------------------------------------------------------
__global__ void final_kernel(const float* __restrict__ dc, float* __restrict__ out) {
    int t = threadIdx.x;
    if (t < TT) {
        float s = 0.f;
        for (int b = 0; b < BB; ++b) s += dc[t * BB + b];
        out[t] = s / (float)BB;
    }
    if (t < BTOT) out[TT + t] = dc[t];
}

extern "C" void kernel_launch(void* const* d_in, const int* in_sizes, int n_in,
                              void* d_out, int out_size, void* d_ws, size_t ws_size,
                              hipStream_t stream) {
    const float* output_points = (const float*)d_in[0];  // [2,4,64,256,3]
    const float* output_range  = (const float*)d_in[1];  // [2,4,64,256]
    const float* target        = (const float*)d_in[2];  // [2,4,4,64,256]
    float* out = (float*)d_out;                          // 4 + 8 floats

    // workspace carve-up (floats)
    float* ws      = (float*)d_ws;
    size_t ptsElems = (size_t)BTOT * NP * 4;
    float4* opPts  = (float4*)ws;                         ws += ptsElems;
    float4* tpPts  = (float4*)ws;                         ws += ptsElems;
    float*  opN    = ws;                                  ws += (size_t)BTOT * NP;
    float*  tpN    = ws;                                  ws += (size_t)BTOT * NP;
    float*  min1   = ws;                                  ws += (size_t)BTOT * NP;
    float*  min2   = ws;                                  ws += (size_t)BTOT * NP;
    float*  dc     = ws;                                  ws += BTOT;

    // 1) prep masked/padded point arrays + norms
    {
        int total = BTOT * NP;
        prep_kernel<<<(total + 255) / 256, 256, 0, stream>>>(
            output_points, output_range, target, opPts, opN, tpPts, tpN);
    }
    // 2) brute-force NN via WMMA: x = row-tile groups (8 waves/block),
    //    y = direction, z = batch*time pair
    {
        dim3 grid((NT + 7) / 8, 2, BTOT);
        nn_kernel<<<grid, 256, 0, stream>>>(opPts, opN, tpPts, tpN, min1, min2);
    }
    // 3) per-pair sums / positive counts -> dist_combined
    reduce_kernel<<<BTOT, 256, 0, stream>>>(min1, min2, dc);
    // 4) 12-element output
    final_kernel<<<1, 32, 0, stream>>>(dc, out);
}